// HANVulClassifier_2499670966293
// MI455X (gfx1250) — compile-verified
//
#include <hip/hip_runtime.h>
#include <math.h>

#define N_NODES   50000
#define D_INF     128
#define HEADS     8
#define FDIM      16
#define HF        128          // HEADS*FDIM
#define M_PATHS   3
#define N_EDGES   300000
#define NUM_GR    16
#define NEG_SLOPE 0.2f

typedef float v2f __attribute__((ext_vector_type(2)));
typedef float v8f __attribute__((ext_vector_type(8)));

// ---------------------------------------------------------------------------
// feat = h @ W^T   (N_NODES x 128) = (N_NODES x 128) @ (128 x 128)^T
// One wave computes a 16-row strip x all 128 cols via V_WMMA_F32_16X16X4_F32.
// A-frag layout (16x4 f32): lane<16: {K=k0,k0+1} of row (row0+lane),
//                           lane>=16: {K=k0+2,k0+3} of row (row0+lane-16).
// B-frag (4x16, B[k][n] = W[n][k]): lane<16: {k0,k0+1} of col n=lane, etc.
// C/D 16x16 f32: vgpr r, lane l: D[r + 8*(l>=16)][l%16].
// ---------------------------------------------------------------------------
__global__ __launch_bounds__(256) void gemm_wmma_f32(
    const float* __restrict__ h, const float* __restrict__ W,
    float* __restrict__ feat)
{
    const int lane = threadIdx.x & 31;
    const int wave = threadIdx.x >> 5;
    const int rt   = blockIdx.x * 8 + wave;     // 16-row tile index
    if (rt * 16 >= N_NODES) return;             // wave-uniform: EXEC stays all-1s
    const int row0 = rt * 16;
    const int half = lane >> 4;                 // 0 or 1
    const int l16  = lane & 15;

    v8f acc[8];
#pragma unroll
    for (int j = 0; j < 8; ++j) acc[j] = v8f{0.f,0.f,0.f,0.f,0.f,0.f,0.f,0.f};

    const float* __restrict__ arow = h + (size_t)(row0 + l16) * D_INF + 2 * half;

    for (int k0 = 0; k0 < D_INF; k0 += 4) {
        const v2f a = *(const v2f*)(arow + k0);
#pragma unroll
        for (int j = 0; j < 8; ++j) {
            const v2f b = *(const v2f*)(W + (size_t)(j * 16 + l16) * D_INF
                                          + 2 * half + k0);
            acc[j] = __builtin_amdgcn_wmma_f32_16x16x4_f32(
                false, a, false, b, (short)0, acc[j], false, false);
        }
    }

#pragma unroll
    for (int j = 0; j < 8; ++j)
#pragma unroll
        for (int r = 0; r < 8; ++r)
            feat[(size_t)(row0 + r + 8 * half) * HF + j * 16 + l16] = acc[j][r];
}

// el[n,h] = sum_f feat[n,h,f]*attn_l[h,f]; er likewise. One thread per (n,h).
__global__ void el_er_kernel(const float* __restrict__ feat,
                             const float* __restrict__ al,
                             const float* __restrict__ ar,
                             float* __restrict__ el, float* __restrict__ er)
{
    int idx = blockIdx.x * blockDim.x + threadIdx.x;
    if (idx >= N_NODES * HEADS) return;
    int node = idx >> 3, head = idx & 7;
    const float* f = feat + (size_t)node * HF + head * FDIM;
    float sl = 0.f, sr = 0.f;
#pragma unroll
    for (int i = 0; i < FDIM; ++i) {
        sl = fmaf(f[i], al[head * FDIM + i], sl);
        sr = fmaf(f[i], ar[head * FDIM + i], sr);
    }
    el[idx] = sl; er[idx] = sr;
}

__global__ void fill_f32(float* __restrict__ p, float v, int n)
{
    int i = blockIdx.x * blockDim.x + threadIdx.x;
    if (i < n) p[i] = v;
}

// Pass A: e = leaky_relu(el[src]+er[dst]); store; atomicMax into emax[dst,h].
__global__ void edge_pass_a(const int* __restrict__ src, const int* __restrict__ dst,
                            const float* __restrict__ el, const float* __restrict__ er,
                            float* __restrict__ e_edge, float* __restrict__ emax)
{
    int e = blockIdx.x * blockDim.x + threadIdx.x;
    if (e >= N_EDGES) return;
    int s = src[e], d = dst[e];
#pragma unroll
    for (int h2 = 0; h2 < HEADS; ++h2) {
        float v = el[s * HEADS + h2] + er[d * HEADS + h2];
        v = v > 0.f ? v : NEG_SLOPE * v;
        e_edge[(size_t)h2 * N_EDGES + e] = v;
        atomicMax(&emax[d * HEADS + h2], v);   // -> global_atomic_max_num_f32
    }
}

// Pass B: ee = exp(e - emax[dst]); store back; native f32 atomic into den.
__global__ void edge_pass_b(const int* __restrict__ dst,
                            float* __restrict__ e_edge,
                            const float* __restrict__ emax,
                            float* __restrict__ den)
{
    int e = blockIdx.x * blockDim.x + threadIdx.x;
    if (e >= N_EDGES) return;
    int d = dst[e];
#pragma unroll
    for (int h2 = 0; h2 < HEADS; ++h2) {
        float ee = expf(e_edge[(size_t)h2 * N_EDGES + e] - emax[d * HEADS + h2]);
        e_edge[(size_t)h2 * N_EDGES + e] = ee;
        unsafeAtomicAdd(&den[d * HEADS + h2], ee);  // -> global_atomic_add_f32
    }
}

// Pass C: rst[dst,h,f] += (ee/den) * feat[src,h,f]. 32 lanes per edge,
// each lane owns 4 consecutive features (all inside one head: h = lane>>2).
__global__ void edge_pass_c(const int* __restrict__ src, const int* __restrict__ dst,
                            const float* __restrict__ e_edge,
                            const float* __restrict__ den,
                            const float* __restrict__ feat,
                            float* __restrict__ rst)
{
    long long gid = (long long)blockIdx.x * blockDim.x + threadIdx.x;
    if (gid >= (long long)N_EDGES * 32) return;
    int e    = (int)(gid >> 5);
    int lane = (int)(gid & 31);
    int c0   = lane * 4;           // feature base 0..124
    int h2   = lane >> 2;          // head for all 4 features
    int s = src[e], d = dst[e];
    float a = e_edge[(size_t)h2 * N_EDGES + e] / den[d * HEADS + h2];
    const float4 fv = *(const float4*)(feat + (size_t)s * HF + c0);
    float* rb = rst + (size_t)d * HF + c0;
    unsafeAtomicAdd(rb + 0, a * fv.x);
    unsafeAtomicAdd(rb + 1, a * fv.y);
    unsafeAtomicAdd(rb + 2, a * fv.z);
    unsafeAtomicAdd(rb + 3, a * fv.w);
}

// z = elu(rst + bias); feats += z/3
__global__ void finalize_m(const float* __restrict__ rst,
                           const float* __restrict__ bias,
                           float* __restrict__ feats)
{
    int idx = blockIdx.x * blockDim.x + threadIdx.x;
    if (idx >= N_NODES * HF) return;
    int c = idx & (HF - 1);
    float z = rst[idx] + bias[c];
    z = z > 0.f ? z : expm1f(z);
    feats[idx] += z * (1.f / 3.f);
}

// Blocked mean pool: LDS partials per graph, then one atomic per (g,c) per block.
#define POOL_NPB 128
__global__ __launch_bounds__(128) void pool_kernel(
    const float* __restrict__ feats, const int* __restrict__ filename,
    float* __restrict__ pooled, float* __restrict__ counts)
{
    __shared__ float lp[NUM_GR * HF];
    __shared__ float lc[NUM_GR];
    int c = threadIdx.x;                       // 0..127
    for (int i = c; i < NUM_GR * HF; i += HF) lp[i] = 0.f;
    if (c < NUM_GR) lc[c] = 0.f;
    __syncthreads();
    int start = blockIdx.x * POOL_NPB;
    int end   = min(start + POOL_NPB, N_NODES);
    for (int n = start; n < end; ++n) {
        int g = filename[n];
        lp[g * HF + c] += feats[(size_t)n * HF + c];
        if (c == 0) lc[g] += 1.f;
    }
    __syncthreads();
    for (int i = c; i < NUM_GR * HF; i += HF) unsafeAtomicAdd(&pooled[i], lp[i]);
    if (c < NUM_GR) unsafeAtomicAdd(&counts[c], lc[c]);
}

// logits[g,o] = dot(pooled[g]/max(count,1), cls_w[o]) + cls_b[o]; 32 lanes.
__global__ void logits_kernel(const float* __restrict__ pooled,
                              const float* __restrict__ counts,
                              const float* __restrict__ cls_w,
                              const float* __restrict__ cls_b,
                              float* __restrict__ out)
{
    int t = threadIdx.x;
    if (t >= NUM_GR * 2) return;
    int g = t >> 1, o = t & 1;
    float inv = 1.f / fmaxf(counts[g], 1.f);
    float s = 0.f;
    for (int c2 = 0; c2 < HF; ++c2)
        s = fmaf(pooled[g * HF + c2] * inv, cls_w[o * HF + c2], s);
    out[g * 2 + o] = s + cls_b[o];
}

// ---------------------------------------------------------------------------
extern "C" void kernel_launch(void* const* d_in, const int* in_sizes, int n_in,
                              void* d_out, int out_size, void* d_ws, size_t ws_size,
                              hipStream_t stream)
{
    const float* h      = (const float*)d_in[0];
    const float* fc_w   = (const float*)d_in[1];   // (M, 128, 128)
    const float* attn_l = (const float*)d_in[2];   // (M, 8, 16)
    const float* attn_r = (const float*)d_in[3];
    const float* gat_b  = (const float*)d_in[4];   // (M, 128)
    // d_in[5..7] (sem_*) are provably dead: softmax over a singleton axis == 1
    const float* cls_w  = (const float*)d_in[8];   // (2, 128)
    const float* cls_b  = (const float*)d_in[9];   // (2,)
    const int*   src    = (const int*)d_in[10];    // (M, E)
    const int*   dst    = (const int*)d_in[11];
    const int*   fname  = (const int*)d_in[12];    // (N,)

    float* out_logits = (float*)d_out;             // 16*2
    float* out_feats  = out_logits + NUM_GR * 2;   // 50000*128

    float* ws     = (float*)d_ws;
    float* feat   = ws;  ws += (size_t)N_NODES * HF;      // 6.4M
    float* el     = ws;  ws += (size_t)N_NODES * HEADS;   // 400K
    float* er     = ws;  ws += (size_t)N_NODES * HEADS;
    float* emax   = ws;  ws += (size_t)N_NODES * HEADS;
    float* den    = ws;  ws += (size_t)N_NODES * HEADS;
    float* e_edge = ws;  ws += (size_t)N_EDGES * HEADS;   // 2.4M
    float* rst    = ws;  ws += (size_t)N_NODES * HF;      // 6.4M
    float* pooled = ws;  ws += NUM_GR * HF;
    float* counts = ws;  ws += NUM_GR;

    const int TB = 256;
    const int nNF = N_NODES * HF;                       // 6,400,000
    const int nNH = N_NODES * HEADS;                    // 400,000

    // Zero accumulators (harness poisons buffers; we own initialization).
    fill_f32<<<(nNF + TB - 1) / TB, TB, 0, stream>>>(out_feats, 0.f, nNF);
    fill_f32<<<(NUM_GR * HF + NUM_GR + TB - 1) / TB, TB, 0, stream>>>(pooled, 0.f,
                                                    NUM_GR * HF + NUM_GR);

    const int gemm_blocks = (N_NODES / 16 + 7) / 8;     // 391
    for (int m = 0; m < M_PATHS; ++m) {
        gemm_wmma_f32<<<gemm_blocks, 256, 0, stream>>>(
            h, fc_w + (size_t)m * HF * D_INF, feat);
        el_er_kernel<<<(nNH + TB - 1) / TB, TB, 0, stream>>>(
            feat, attn_l + m * HF, attn_r + m * HF, el, er);
        fill_f32<<<(nNH + TB - 1) / TB, TB, 0, stream>>>(emax, -INFINITY, nNH);
        fill_f32<<<(nNH + TB - 1) / TB, TB, 0, stream>>>(den, 0.f, nNH);
        fill_f32<<<(nNF + TB - 1) / TB, TB, 0, stream>>>(rst, 0.f, nNF);

        const int* sm = src + (size_t)m * N_EDGES;
        const int* dm = dst + (size_t)m * N_EDGES;
        edge_pass_a<<<(N_EDGES + TB - 1) / TB, TB, 0, stream>>>(
            sm, dm, el, er, e_edge, emax);
        edge_pass_b<<<(N_EDGES + TB - 1) / TB, TB, 0, stream>>>(
            dm, e_edge, emax, den);
        long long cthreads = (long long)N_EDGES * 32;
        edge_pass_c<<<(int)((cthreads + TB - 1) / TB), TB, 0, stream>>>(
            sm, dm, e_edge, den, feat, rst);

        finalize_m<<<(nNF + TB - 1) / TB, TB, 0, stream>>>(
            rst, gat_b + m * HF, out_feats);
    }

    pool_kernel<<<(N_NODES + POOL_NPB - 1) / POOL_NPB, 128, 0, stream>>>(
        out_feats, fname, pooled, counts);
    logits_kernel<<<1, 32, 0, stream>>>(pooled, counts, cls_w, cls_b, out_logits);
}